// GNN_v7_5927054868950
// MI455X (gfx1250) — compile-verified
//
#include <hip/hip_runtime.h>
#include <stdint.h>

// f32 WMMA fragment types for gfx1250 (wave32):
// A 16x4 f32  -> 64 vals / 32 lanes = 2 VGPRs
// B 4x16 f32  -> 2 VGPRs
// C/D 16x16   -> 256 vals / 32 lanes = 8 VGPRs
typedef __attribute__((ext_vector_type(2))) float v2f;
typedef __attribute__((ext_vector_type(8))) float v8f;

// Global (address_space(1)) pointer: weights are global memory; forcing AS1
// turns flat_load_b32 (which also ties up DScnt) into pure global_load_b32.
typedef const float __attribute__((address_space(1))) gfloat;

#define N_NODES 7
#define NE      14
#define DCH     128
#define M_PAD   16

__global__ __launch_bounds__(256) void gnn_fused_kernel(
    const float* __restrict__ x,
    const float* __restrict__ lep_W1, const float* __restrict__ lep_b1,
    const float* __restrict__ lep_W2, const float* __restrict__ lep_b2,
    const float* __restrict__ me_W1,  const float* __restrict__ me_b1,
    const float* __restrict__ me_W2,  const float* __restrict__ me_b2,
    const float* __restrict__ jet_W1, const float* __restrict__ jet_b1,
    const float* __restrict__ jet_W2, const float* __restrict__ jet_b2,
    const float* __restrict__ hl_W1,  const float* __restrict__ hl_b1,
    const float* __restrict__ hl_W2,  const float* __restrict__ hl_b2,
    const float* __restrict__ a1_Wi,  const float* __restrict__ a1_Wr, const float* __restrict__ a1_b,
    const float* __restrict__ a2_Wi,  const float* __restrict__ a2_Wr, const float* __restrict__ a2_b,
    const float* __restrict__ cls_W1, const float* __restrict__ cls_b1,
    const float* __restrict__ cls_W2, const float* __restrict__ cls_b2,
    const int* __restrict__ edge_index,
    float* __restrict__ out)
{
    __shared__ float node[M_PAD][DCH];   // activation matrix, rows 7..15 = 0 padding
    __shared__ float hWi [M_PAD][DCH];   // node @ Wi
    __shared__ float hWr [M_PAD][DCH];   // node @ Wr
    __shared__ float hbuf[64];           // MLP hidden vector
    __shared__ float pooled[DCH];
    __shared__ float norm_s[NE];
    __shared__ int   src_s[NE];
    __shared__ int   dst_s[NE];

    const int tid  = threadIdx.x;
    const int lane = tid & 31;
    const int wave = tid >> 5;

    // ---- prefetch the 4x64KB ARMA weight matrices (global_prefetch_b8) ----
    // Overlaps HBM/L2 fill with the serial node-MLP phase below.
    // 256 threads x 256 bytes == 64KB == one full 128x128 f32 matrix each.
    {
        const size_t poff = (size_t)tid * 256;
        __builtin_prefetch((const char*)a1_Wi + poff, 0, 3);
        __builtin_prefetch((const char*)a1_Wr + poff, 0, 3);
        __builtin_prefetch((const char*)a2_Wi + poff, 0, 3);
        __builtin_prefetch((const char*)a2_Wr + poff, 0, 3);
    }

    // ---- GCN edge norms (tiny fixed graph: 14 edges, 7 nodes) ----
    if (tid == 0) {
        float deg[N_NODES];
        #pragma unroll
        for (int i = 0; i < N_NODES; ++i) deg[i] = 0.0f;
        for (int e = 0; e < NE; ++e) {
            src_s[e] = edge_index[e];
            dst_s[e] = edge_index[NE + e];
            deg[dst_s[e]] += 1.0f;
        }
        float dinv[N_NODES];
        #pragma unroll
        for (int i = 0; i < N_NODES; ++i)
            dinv[i] = (deg[i] > 0.0f) ? rsqrtf(deg[i]) : 0.0f;
        for (int e = 0; e < NE; ++e)
            norm_s[e] = dinv[src_s[e]] * dinv[dst_s[e]];
    }

    // zero the padding rows of the activation matrix
    for (int i = tid; i < (M_PAD - N_NODES) * DCH; i += 256)
        node[N_NODES + i / DCH][i % DCH] = 0.0f;
    __syncthreads();

    // ---- node feature MLPs: only x[0] feeds the output (stacked[0]) ----
    const float* W1s[N_NODES] = {lep_W1, me_W1, jet_W1, jet_W1, jet_W1, jet_W1, hl_W1};
    const float* b1s[N_NODES] = {lep_b1, me_b1, jet_b1, jet_b1, jet_b1, jet_b1, hl_b1};
    const float* W2s[N_NODES] = {lep_W2, me_W2, jet_W2, jet_W2, jet_W2, jet_W2, hl_W2};
    const float* b2s[N_NODES] = {lep_b2, me_b2, jet_b2, jet_b2, jet_b2, jet_b2, hl_b2};
    const int din[N_NODES] = {3, 2, 4, 4, 4, 4, 7};
    const int off[N_NODES] = {0, 3, 5, 9, 13, 17, 21};

    for (int n = 0; n < N_NODES; ++n) {
        if (tid < 64) {                               // hidden: din -> 64, relu
            float acc = b1s[n][tid];
            for (int i = 0; i < din[n]; ++i)
                acc = fmaf(x[off[n] + i], W1s[n][i * 64 + tid], acc);
            hbuf[tid] = fmaxf(acc, 0.0f);
        }
        __syncthreads();
        if (tid < DCH) {                              // out: 64 -> 128, linear
            float acc = b2s[n][tid];
            for (int k = 0; k < 64; ++k)
                acc = fmaf(hbuf[k], W2s[n][k * DCH + tid], acc);
            node[n][tid] = acc;
        }
        __syncthreads();
    }

    // ---- [16,128] x [128,128] GEMM via V_WMMA_F32_16X16X4_F32 ----
    // wave w owns output columns [16w, 16w+16). Uniform control flow: EXEC all-ones.
    auto gemm_tile = [&](const float (*A)[DCH], const float* Bp,
                         float (*Out)[DCH]) {
        gfloat* Bg = (gfloat*)(unsigned long long)(uintptr_t)Bp;  // force AS1
        const int m     = lane & 15;            // A row / B,C column index within tile
        const int khalf = (lane >> 4) << 1;     // lanes 0-15: K=0,1 ; lanes 16-31: K=2,3
        const int col   = (wave << 4) + m;      // global output column
        v8f c = {};
        #pragma unroll 4
        for (int k0 = 0; k0 < DCH; k0 += 4) {
            v2f a, b;
            a.x = A[m][k0 + khalf];
            a.y = A[m][k0 + khalf + 1];
            b.x = Bg[(k0 + khalf)     * DCH + col];   // coalesced across lanes
            b.y = Bg[(k0 + khalf + 1) * DCH + col];
            c = __builtin_amdgcn_wmma_f32_16x16x4_f32(
                    /*neg_a=*/false, a, /*neg_b=*/false, b,
                    /*c_mod=*/(short)0, c, /*reuse_a=*/false, /*reuse_b=*/false);
        }
        // C/D layout: lanes 0-15 hold M=0..7 in VGPRs 0..7, lanes 16-31 hold M=8..15
        const int rbase = (lane >> 4) << 3;
        #pragma unroll
        for (int r = 0; r < 8; ++r)
            Out[rbase + r][col] = c[r];
    };

    // ---- ARMAConv layer: relu( A_norm (node Wi) + node Wr + b ) ----
    auto arma = [&](const float* Wi, const float* Wr, const float* b) {
        gemm_tile(node, Wi, hWi);
        gemm_tile(node, Wr, hWr);
        __syncthreads();
        if (tid < DCH) {
            #pragma unroll
            for (int v = 0; v < N_NODES; ++v) {
                float acc = hWr[v][tid] + b[tid];
                for (int e = 0; e < NE; ++e)
                    if (dst_s[e] == v)
                        acc = fmaf(norm_s[e], hWi[src_s[e]][tid], acc);
                node[v][tid] = fmaxf(acc, 0.0f);
            }
        }
        __syncthreads();
    };

    arma(a1_Wi, a1_Wr, a1_b);
    arma(a2_Wi, a2_Wr, a2_b);

    // ---- global max pool over the 7 real nodes ----
    if (tid < DCH) {
        float m = node[0][tid];
        #pragma unroll
        for (int v = 1; v < N_NODES; ++v) m = fmaxf(m, node[v][tid]);
        pooled[tid] = m;
    }
    __syncthreads();

    // ---- classifier MLP: 128 -> 64 (relu) -> 1 ----
    if (tid < 64) {
        float acc = cls_b1[tid];
        for (int c = 0; c < DCH; ++c)
            acc = fmaf(pooled[c], cls_W1[c * 64 + tid], acc);
        hbuf[tid] = fmaxf(acc, 0.0f);
    }
    __syncthreads();
    if (tid == 0) {
        float acc = cls_b2[0];
        for (int k = 0; k < 64; ++k)
            acc = fmaf(hbuf[k], cls_W2[k], acc);
        out[0] = acc;
    }
}

extern "C" void kernel_launch(void* const* d_in, const int* in_sizes, int n_in,
                              void* d_out, int out_size, void* d_ws, size_t ws_size,
                              hipStream_t stream) {
    (void)in_sizes; (void)n_in; (void)out_size; (void)d_ws; (void)ws_size;
    const float* x      = (const float*)d_in[0];
    const float* lep_W1 = (const float*)d_in[1];
    const float* lep_b1 = (const float*)d_in[2];
    const float* lep_W2 = (const float*)d_in[3];
    const float* lep_b2 = (const float*)d_in[4];
    const float* me_W1  = (const float*)d_in[5];
    const float* me_b1  = (const float*)d_in[6];
    const float* me_W2  = (const float*)d_in[7];
    const float* me_b2  = (const float*)d_in[8];
    const float* jet_W1 = (const float*)d_in[9];
    const float* jet_b1 = (const float*)d_in[10];
    const float* jet_W2 = (const float*)d_in[11];
    const float* jet_b2 = (const float*)d_in[12];
    const float* hl_W1  = (const float*)d_in[13];
    const float* hl_b1  = (const float*)d_in[14];
    const float* hl_W2  = (const float*)d_in[15];
    const float* hl_b2  = (const float*)d_in[16];
    const float* a1_Wi  = (const float*)d_in[17];
    const float* a1_Wr  = (const float*)d_in[18];
    const float* a1_b   = (const float*)d_in[19];
    const float* a2_Wi  = (const float*)d_in[20];
    const float* a2_Wr  = (const float*)d_in[21];
    const float* a2_b   = (const float*)d_in[22];
    const float* cls_W1 = (const float*)d_in[23];
    const float* cls_b1 = (const float*)d_in[24];
    const float* cls_W2 = (const float*)d_in[25];
    const float* cls_b2 = (const float*)d_in[26];
    const int*   eidx   = (const int*)d_in[27];
    float* out = (float*)d_out;

    // Entire pipeline is one tiny fused workgroup: the output depends only on
    // x[0] (stacked[0] in the reference), so the 262144-row batch is dead code.
    gnn_fused_kernel<<<1, 256, 0, stream>>>(
        x, lep_W1, lep_b1, lep_W2, lep_b2, me_W1, me_b1, me_W2, me_b2,
        jet_W1, jet_b1, jet_W2, jet_b2, hl_W1, hl_b1, hl_W2, hl_b2,
        a1_Wi, a1_Wr, a1_b, a2_Wi, a2_Wr, a2_b,
        cls_W1, cls_b1, cls_W2, cls_b2, eidx, out);
}